// MaskedCNN_70746701299893
// MI455X (gfx1250) — compile-verified
//
#include <hip/hip_runtime.h>

typedef float v2f __attribute__((ext_vector_type(2)));
typedef float v8f __attribute__((ext_vector_type(8)));
typedef unsigned int u32x4 __attribute__((ext_vector_type(4)));
typedef unsigned int u32x8 __attribute__((ext_vector_type(8)));

// ---------------------------------------------------------------- init ------
__global__ void k_init(float* __restrict__ psum,
                       unsigned long long* __restrict__ pmax,
                       float* __restrict__ dense_acc) {
    int t = threadIdx.x + blockIdx.x * blockDim.x;        // 32768 threads
    if (t < 32 * 64) { psum[t] = 0.f; pmax[t] = 0ull; }
    if (t < 32 * 1024) dense_acc[t] = 0.f;
}

// ------------------------------------------- conv1 + relu + maxpool2 --------
// in: (32,128,128,1)  w:(3,3,1,32)  -> pool1 (32,64,64,32)
__global__ void k_conv1pool(const float* __restrict__ in,
                            const float* __restrict__ w,
                            const float* __restrict__ bias,
                            float* __restrict__ pool1) {
    int t  = threadIdx.x + blockIdx.x * blockDim.x;       // 32*64*64*32
    int ch = t & 31;
    int x  = (t >> 5) & 63;
    int y  = (t >> 11) & 63;
    int b  = t >> 17;
    float wk[9];
#pragma unroll
    for (int g = 0; g < 9; ++g) wk[g] = w[g * 32 + ch];
    float bb = bias[ch];
    float m = 0.f;                                        // post-relu values >= 0
#pragma unroll
    for (int dy = 0; dy < 2; ++dy)
#pragma unroll
    for (int dx = 0; dx < 2; ++dx) {
        int Y = 2 * y + dy, X = 2 * x + dx;
        float acc = bb;
#pragma unroll
        for (int ky = 0; ky < 3; ++ky)
#pragma unroll
        for (int kx = 0; kx < 3; ++kx) {
            int iy = Y + ky - 1, ix = X + kx - 1;
            float v = 0.f;
            if (iy >= 0 && iy < 128 && ix >= 0 && ix < 128)
                v = in[((size_t)b * 128 + iy) * 128 + ix];
            acc = fmaf(v, wk[ky * 3 + kx], acc);
        }
        m = fmaxf(m, fmaxf(acc, 0.f));
    }
    pool1[t] = m;   // layout ((b*64+y)*64+x)*32+ch == t
}

// ---------------------- conv2 via f32 WMMA + stats (sum, packed argmax) -----
// pool1 (32,64,64,32) -> conv2 (32,64,64,64), relu fused.
// Per WG: batch b, 16x16 spatial tile. GEMM: M=256 pixels, K=288, N=64.
// Weights (73728 B contiguous, K-major) are DMA'd into LDS by the TDM while
// the VALU lanes fill the halo patch.
__global__ __launch_bounds__(256)
void k_conv2(const float* __restrict__ pool1, const float* __restrict__ w2,
             const float* __restrict__ b2, float* __restrict__ conv2,
             float* __restrict__ psum, unsigned long long* __restrict__ pmax) {
    __shared__ float wlds[288 * 64];                      // 72 KB: all weights
    __shared__ float patch[18 * 18 * 32];                 // 41 KB: halo patch
    __shared__ float csum[64];
    __shared__ unsigned long long cmax[64];

    const int tid = threadIdx.x;
    const int b   = blockIdx.y;
    const int ty  = blockIdx.x >> 2, tx = blockIdx.x & 3;
    const int wave = tid >> 5, lane = tid & 31;

    // ---- TDM: tensor_load_to_lds of the whole weight tensor (1D tile) ----
    if (wave == 0) {
        unsigned long long ga = (unsigned long long)(const void*)w2;
        unsigned ldsOff = (unsigned)(unsigned long long)(void*)wlds; // LDS byte offset
        // D# group0: count=1 | lds_addr | global_addr[56:0] | type=2
        u32x4 g0 = { 1u, ldsOff, (unsigned)ga,
                     ((unsigned)(ga >> 32) & 0x01FFFFFFu) | 0x80000000u };
        // D# group1: data_size=4B, tensor_dim0=tile_dim0=18432, dim1 unused
        u32x8 g1 = { 0x00020000u,            // mask=0, data_size=2 (4B)
                     0x48000000u,            // tensor_dim0[15:0]=18432 <<16
                     0x00010000u,            // tensor_dim1=1
                     0x48000000u,            // tile_dim0=18432 <<16
                     0u,                     // tile_dim1=0, tile_dim2=0
                     18432u,                 // tensor_dim0_stride[31:0]
                     0u, 0u };
        u32x4 gz = { 0u, 0u, 0u, 0u };       // groups 2/3: dims 3/4 unused
        asm volatile("tensor_load_to_lds %0, %1, %2, %3"
                     :: "s"(g0), "s"(g1), "s"(gz), "s"(gz)
                     : "memory");
    }

    const int py0 = ty * 16 - 1, px0 = tx * 16 - 1;
    for (int i = tid; i < 18 * 18 * 32; i += 256) {
        int ci = i & 31, r = i >> 5;
        int xx = r % 18, yy = r / 18;
        int y = py0 + yy, x = px0 + xx;
        float v = 0.f;
        if (y >= 0 && y < 64 && x >= 0 && x < 64)
            v = pool1[(((size_t)b * 64 + y) * 64 + x) * 32 + ci];
        patch[i] = v;
    }
    if (tid < 64) { csum[tid] = 0.f; cmax[tid] = 0ull; }
    if (wave == 0) __builtin_amdgcn_s_wait_tensorcnt(0);  // TDM done (wave 0)
    __syncthreads();                                      // publish wlds + patch

    const int ml = lane & 15, hi = lane >> 4;

    for (int i = 0; i < 8; ++i) {                         // 8 (mt,nt) tiles/wave
        int combo = wave + 8 * i;
        int mt = combo >> 2, nt = combo & 3;              // mt = tile row, 16 px
        int n  = nt * 16 + ml;                            // output channel (B/C lane)
        v8f acc = {};
        for (int k = 0; k < 288; k += 4) {
            int kk = k + 2 * hi;                          // per-lane K pair start
            int g = kk >> 5, ci = kk & 31;                // tap, input channel
            int ky = g / 3, kx = g % 3;
            // A: row M=ml -> pixel (py=mt, px=ml); two consecutive ci -> b64 LDS load
            v2f a = *(const v2f*)&patch[((mt + ky) * 18 + (ml + kx)) * 32 + ci];
            v2f bf;
            bf.x = wlds[kk * 64 + n];
            bf.y = wlds[kk * 64 + 64 + n];
            acc = __builtin_amdgcn_wmma_f32_16x16x4_f32(
                false, a, false, bf, (short)0, acc, false, false);
        }
        float bb = b2[n];
        float localsum = 0.f;
        unsigned long long localmax = 0ull;
        int yy = ty * 16 + mt;
        int sbase = yy * 64 + tx * 16 + 8 * hi;           // spatial idx of r=0
        float* dst = conv2 + (((size_t)b * 64 + yy) * 64 + tx * 16 + 8 * hi) * 64 + n;
#pragma unroll
        for (int r = 0; r < 8; ++r) {                     // C: M = r + 8*hi -> xx += r
            float v = fmaxf(acc[r] + bb, 0.f);
            dst[r * 64] = v;                              // xx step == 64 floats
            localsum += v;
            unsigned long long p =
                ((unsigned long long)__float_as_uint(v) << 32) |
                (unsigned int)(~(unsigned int)(sbase + r));
            localmax = p > localmax ? p : localmax;
        }
        atomicAdd(&csum[n], localsum);                    // ds_add_f32
        atomicMax(&cmax[n], localmax);                    // ds_max_u64
    }
    __syncthreads();
    if (tid < 64) {
        atomicAdd(&psum[b * 64 + tid], csum[tid]);
        atomicMax(&pmax[b * 64 + tid], cmax[tid]);
    }
}

// ----------------- mask from argmax + masked relu + maxpool2 -> d_out -------
__global__ void k_maskpool(const float* __restrict__ conv2,
                           const unsigned long long* __restrict__ pmax,
                           float* __restrict__ out) {
    int t  = threadIdx.x + blockIdx.x * blockDim.x;       // 32*32*32*64
    int ch = t & 63;
    int x  = (t >> 6) & 31;
    int y  = (t >> 11) & 31;
    int b  = t >> 16;
    unsigned int s = (~(unsigned int)pmax[b * 64 + ch]) & 4095u;
    float mux = (float)(s >> 6) * (1.f / 31.5f) - 1.f;
    float muy = (float)(s & 63) * (1.f / 31.5f) - 1.f;
    float m = 0.f;
#pragma unroll
    for (int dy = 0; dy < 2; ++dy)
#pragma unroll
    for (int dx = 0; dx < 2; ++dx) {
        int yy = 2 * y + dy, xx = 2 * x + dx;
        float liny = -1.f + (2.f / 63.f) * (float)yy;
        float linx = -1.f + (2.f / 63.f) * (float)xx;
        float mask = fmaxf(1.f - 0.5f * (fabsf(liny - mux) + fabsf(linx - muy)), -1.f);
        float v = conv2[(((size_t)b * 64 + yy) * 64 + xx) * 64 + ch] * mask;
        m = fmaxf(m, fmaxf(v, 0.f));
    }
    out[t] = m;   // ((b*32+y)*32+x)*64+ch == flattened dense input layout
}

// -------------------- dense1: (32x65536)@(65536x1024) via f32 WMMA ----------
// grid (64 Ntiles, 4 Ksplits); 8 waves partition K inside a WG; LDS reduce;
// partial tiles combined via atomicAdd into dense_acc (bias/relu fused later).
__global__ __launch_bounds__(256)
void k_dense1(const float* __restrict__ A, const float* __restrict__ W,
              float* __restrict__ acc_out) {
    __shared__ float red[16 * 256];                       // [tile*8+r][wave*32+lane]
    const int tid = threadIdx.x, wave = tid >> 5, lane = tid & 31;
    const int ml = lane & 15, hi = lane >> 4;
    const int n0 = blockIdx.x * 16;
    const int kbase = blockIdx.y * 16384 + wave * 4;

    const float* a0p = A + (size_t)ml * 65536 + kbase + 2 * hi;
    const float* a1p = a0p + (size_t)16 * 65536;
    const float* bp  = W + (size_t)(kbase + 2 * hi) * 1024 + n0 + ml;

    v8f c0 = {}, c1 = {};
    for (int it = 0; it < 512; ++it) {                    // 16384 K / (8 waves * 4)
        v2f a0 = *(const v2f*)a0p;                        // A rows 0..15
        v2f a1 = *(const v2f*)a1p;                        // A rows 16..31
        v2f bf;
        bf.x = bp[0];
        bf.y = bp[1024];
        __builtin_prefetch(bp + 8 * 32 * 1024, 0, 0);     // stream weights ahead
        c0 = __builtin_amdgcn_wmma_f32_16x16x4_f32(
            false, a0, false, bf, (short)0, c0, false, false);
        c1 = __builtin_amdgcn_wmma_f32_16x16x4_f32(
            false, a1, false, bf, (short)0, c1, false, false);
        a0p += 32; a1p += 32; bp += 32 * 1024;
    }
#pragma unroll
    for (int r = 0; r < 8; ++r) {
        red[(r)      * 256 + wave * 32 + lane] = c0[r];
        red[(8 + r)  * 256 + wave * 32 + lane] = c1[r];
    }
    __syncthreads();
    for (int idx = tid; idx < 512; idx += 256) {
        int slot = idx >> 5, ln = idx & 31;
        float s = 0.f;
#pragma unroll
        for (int w8 = 0; w8 < 8; ++w8) s += red[slot * 256 + w8 * 32 + ln];
        int tile = slot >> 3, r = slot & 7;
        int mrow = tile * 16 + r + 8 * (ln >> 4);
        int n = n0 + (ln & 15);
        atomicAdd(&acc_out[mrow * 1024 + n], s);
    }
}

// -------------------- logits (fuses dense1 bias+relu on read) ---------------
__global__ void k_logits(const float* __restrict__ dense_acc,
                         const float* __restrict__ d1b,
                         const float* __restrict__ W2,
                         const float* __restrict__ b2,
                         float* __restrict__ logits) {
    int t = threadIdx.x;                                  // 320
    int b = t / 10, j = t % 10;
    float acc = b2[j];
    for (int k = 0; k < 1024; ++k) {
        float h = fmaxf(dense_acc[b * 1024 + k] + d1b[k], 0.f);
        acc = fmaf(h, W2[k * 10 + j], acc);
    }
    logits[t] = acc;
}

// -------------------- per-category average + argmax -------------------------
__global__ void k_filtercat(const float* __restrict__ psum,
                            const int* __restrict__ labels,
                            float* __restrict__ outfc) {
    int c = threadIdx.x;                                  // 64
    float sums[10]; int cnt[10];
#pragma unroll
    for (int k = 0; k < 10; ++k) { sums[k] = 0.f; cnt[k] = 0; }
    for (int b = 0; b < 32; ++b) {
        int lab = labels[b];
        sums[lab] += psum[b * 64 + c];
        cnt[lab]++;
    }
    float best = sums[0] / (float)cnt[0];
    int bi = 0;
    for (int k = 1; k < 10; ++k) {
        float a = sums[k] / (float)cnt[k];
        if (a > best) { best = a; bi = k; }
    }
    outfc[c] = (float)bi;
}

// ---------------------------------------------------------------------------
extern "C" void kernel_launch(void* const* d_in, const int* in_sizes, int n_in,
                              void* d_out, int out_size, void* d_ws, size_t ws_size,
                              hipStream_t stream) {
    const float* inputs   = (const float*)d_in[0];
    const int*   labels   = (const int*)d_in[1];
    const float* conv1_w  = (const float*)d_in[2];
    const float* conv1_b  = (const float*)d_in[3];
    const float* conv2_w  = (const float*)d_in[4];
    const float* conv2_b  = (const float*)d_in[5];
    const float* dense1_w = (const float*)d_in[6];
    const float* dense1_b = (const float*)d_in[7];
    const float* dense2_w = (const float*)d_in[8];
    const float* dense2_b = (const float*)d_in[9];

    float* out = (float*)d_out;
    char*  ws  = (char*)d_ws;

    float*              pool1     = (float*)(ws);                         // 16 MB
    float*              conv2     = (float*)(ws + 16777216);              // 33.5 MB
    float*              psum      = (float*)(ws + 50331648);              // 8 KB
    unsigned long long* pmax      = (unsigned long long*)(ws + 50339840); // 16 KB
    float*              dense_acc = (float*)(ws + 50356224);              // 128 KB

    float* logits = out;
    float* pool2  = out + 320;
    float* fc     = out + 320 + 2097152;

    k_init<<<128, 256, 0, stream>>>(psum, pmax, dense_acc);
    k_conv1pool<<<16384, 256, 0, stream>>>(inputs, conv1_w, conv1_b, pool1);
    k_conv2<<<dim3(16, 32), 256, 0, stream>>>(pool1, conv2_w, conv2_b,
                                              conv2, psum, pmax);
    k_maskpool<<<8192, 256, 0, stream>>>(conv2, pmax, pool2);
    k_dense1<<<dim3(64, 4), 256, 0, stream>>>(pool2, dense1_w, dense_acc);
    k_logits<<<1, 320, 0, stream>>>(dense_acc, dense1_b, dense2_w, dense2_b, logits);
    k_filtercat<<<1, 64, 0, stream>>>(psum, labels, fc);
}